// HiddenMarkovModel_81698867905098
// MI455X (gfx1250) — compile-verified
//
#include <hip/hip_runtime.h>
#include <math.h>

// HMM forward algorithm, B=64, T=4096, D=90, K=32, fp32 throughout.
typedef __attribute__((ext_vector_type(2))) float v2f;
typedef __attribute__((ext_vector_type(8))) float v8f;

#define HMM_B 64
#define HMM_T 4096
#define HMM_D 90
#define HMM_K 32
#define LOG2PI_F 1.8378770664093453f

// ---- workspace layout (float offsets) ----
#define OFF_W1    0                       // 32*90   inv_var
#define OFF_W2    (32*90)                 // 32*90   -2*mu*inv_var
#define OFF_BIAS  (2*32*90)               // 32      sum(mu^2*iv)+sum(log_var)+D*log2pi
#define OFF_AEXP  (2*32*90 + 32)          // 32*32   row-softmax of transition matrix
#define OFF_LOGPI (2*32*90 + 32 + 1024)   // 32
#define OFF_RES   (2*32*90 + 32 + 1024 + 32) // 64 per-batch results
#define OFF_EM    8192                    // B*T*K floats (32 MB), (B,T,K) layout

__device__ __forceinline__ float bcast_lane(float v, int lane) {
    return __int_as_float(__builtin_amdgcn_readlane(__float_as_int(v), lane));
}

// ---------------------------------------------------------------------------
// Kernel 1: precompute log_pi, softmax(transition) rows, Gaussian weights.
// One wave, lane k = state k.
// ---------------------------------------------------------------------------
__global__ __launch_bounds__(32)
void hmm_prep(const float* __restrict__ sp, const float* __restrict__ trans,
              const float* __restrict__ mu, const float* __restrict__ lv,
              float* __restrict__ ws) {
    const int k = threadIdx.x;

    // log_softmax of state priors (wave32 reductions)
    float s = sp[k];
    float m = s;
    #pragma unroll
    for (int off = 16; off; off >>= 1) m = fmaxf(m, __shfl_xor(m, off, 32));
    float sum = __expf(s - m);
    #pragma unroll
    for (int off = 16; off; off >>= 1) sum += __shfl_xor(sum, off, 32);
    ws[OFF_LOGPI + k] = s - m - __logf(sum);

    // softmax of transition row k (linear domain: exp(log_softmax))
    float tm = -INFINITY;
    for (int j = 0; j < HMM_K; j++) tm = fmaxf(tm, trans[k * HMM_K + j]);
    float ts = 0.f;
    for (int j = 0; j < HMM_K; j++) ts += __expf(trans[k * HMM_K + j] - tm);
    float inv = 1.0f / ts;
    for (int j = 0; j < HMM_K; j++)
        ws[OFF_AEXP + k * HMM_K + j] = __expf(trans[k * HMM_K + j] - tm) * inv;

    // Gaussian emission weights for state k
    float c2 = 0.f, slv = 0.f;
    for (int d = 0; d < HMM_D; d++) {
        float l  = lv[k * HMM_D + d];
        float iv = __expf(-l);
        float mm = mu[k * HMM_D + d];
        ws[OFF_W1 + k * HMM_D + d] = iv;
        ws[OFF_W2 + k * HMM_D + d] = -2.0f * mm * iv;
        c2  += mm * mm * iv;
        slv += l;
    }
    ws[OFF_BIAS + k] = c2 + slv + (float)HMM_D * LOG2PI_F;
}

// ---------------------------------------------------------------------------
// Kernel 2: emission log-probs via V_WMMA_F32_16X16X4_F32.
// Each wave owns a 16-row tile of X (M=16), N=32 states as two 16-wide tiles,
// K-loop over D=90 in chunks of 4 (tail chunk zero-padded branchlessly).
// em[r,k] = -0.5*( sum_d x^2*W1 + x*W2 + bias[k] )
// Memory-bound: 94 MB of X once @ 23.3 TB/s; fp32 WMMA keeps reference precision.
// ---------------------------------------------------------------------------
__global__ __launch_bounds__(256)
void hmm_emission(const float* __restrict__ X, const float* __restrict__ ws,
                  float* __restrict__ em) {
    const int lane = threadIdx.x & 31;
    const int wave = threadIdx.x >> 5;
    const int tile = blockIdx.x * 8 + wave;     // 16-row M tile
    const int rbase = tile * 16;
    const int mrow  = lane & 15;                // A-matrix: lanes 0-15 rows 0-15
    const int hi    = lane >> 4;                // A/B: hi lanes hold K=2,3

    const float* __restrict__ xrow = X + (size_t)(rbase + mrow) * HMM_D;
    const float* __restrict__ w1a = ws + OFF_W1 + (lane & 15) * HMM_D;        // states 0..15
    const float* __restrict__ w1b = ws + OFF_W1 + (16 + (lane & 15)) * HMM_D; // states 16..31
    const float* __restrict__ w2a = ws + OFF_W2 + (lane & 15) * HMM_D;
    const float* __restrict__ w2b = ws + OFF_W2 + (16 + (lane & 15)) * HMM_D;

    v8f c0 = {};   // states 0..15
    v8f c1 = {};   // states 16..31

    #pragma unroll
    for (int d0 = 0; d0 < HMM_D; d0 += 4) {
        const int d  = d0 + 2 * hi;
        const bool ok = (d + 1) < HMM_D;      // false only for hi lanes of tail chunk
        const int dc = ok ? d : 0;            // clamp: keep loads in-bounds, EXEC uniform

        // A operand: x features for this 16x4 chunk (zeroed on padded columns)
        v2f ax;
        ax.x = ok ? xrow[dc]     : 0.0f;
        ax.y = ok ? xrow[dc + 1] : 0.0f;
        v2f ax2 = ax * ax;

        // B operands: weight rows (garbage on padded rows is multiplied by zero A)
        v2f b10, b11, b20, b21;
        b10.x = w1a[dc]; b10.y = w1a[dc + 1];
        b11.x = w1b[dc]; b11.y = w1b[dc + 1];
        b20.x = w2a[dc]; b20.y = w2a[dc + 1];
        b21.x = w2b[dc]; b21.y = w2b[dc + 1];

        c0 = __builtin_amdgcn_wmma_f32_16x16x4_f32(false, ax2, false, b10,
                                                   (short)0, c0, false, false);
        c1 = __builtin_amdgcn_wmma_f32_16x16x4_f32(false, ax2, false, b11,
                                                   (short)0, c1, false, false);
        c0 = __builtin_amdgcn_wmma_f32_16x16x4_f32(false, ax,  false, b20,
                                                   (short)0, c0, false, false);
        c1 = __builtin_amdgcn_wmma_f32_16x16x4_f32(false, ax,  false, b21,
                                                   (short)0, c1, false, false);
    }

    const float bn0 = ws[OFF_BIAS + (lane & 15)];
    const float bn1 = ws[OFF_BIAS + 16 + (lane & 15)];

    // C/D layout: lane holds column n=lane&15, VGPR v holds row v + 8*hi
    #pragma unroll
    for (int v = 0; v < 8; v++) {
        const int r = rbase + v + 8 * hi;
        em[(size_t)r * HMM_K + (lane & 15)]      = -0.5f * (c0[v] + bn0);
        em[(size_t)r * HMM_K + 16 + (lane & 15)] = -0.5f * (c1[v] + bn1);
    }
}

// ---------------------------------------------------------------------------
// Kernel 3: forward recursion. One wave per batch, lane j = state j.
// Step: m = max(alpha); p = exp(alpha-m); v_j = sum_k p_k * A[k][j];
//       alpha_j = e_j + m + log(v_j).
// Latency-optimized: broadcasts via v_readlane (SGPR operand into v_fmac),
// 4-way split FMA accumulation, em row software-pipelined (L2-resident).
// ---------------------------------------------------------------------------
__global__ __launch_bounds__(32)
void hmm_scan(const float* __restrict__ ws, const float* __restrict__ em,
              float* __restrict__ res) {
    const int b = blockIdx.x;
    const int j = threadIdx.x;

    float Acol[HMM_K];
    #pragma unroll
    for (int k = 0; k < HMM_K; k++) Acol[k] = ws[OFF_AEXP + k * HMM_K + j];

    const float* __restrict__ embt = em + (size_t)b * HMM_T * HMM_K;
    float alpha  = ws[OFF_LOGPI + j] + embt[j];
    float e_next = embt[HMM_K + j];              // t = 1 row, pipelined

    for (int t = 1; t < HMM_T; t++) {
        const float e = e_next;
        const int tn = (t + 1 < HMM_T) ? (t + 1) : t;   // clamp, branchless
        e_next = embt[tn * HMM_K + j];                  // hits L2 (em is resident)

        float m = alpha;
        #pragma unroll
        for (int off = 16; off; off >>= 1) m = fmaxf(m, __shfl_xor(m, off, 32));
        const float p = __expf(alpha - m);

        float v0 = 0.f, v1 = 0.f, v2 = 0.f, v3 = 0.f;
        #pragma unroll
        for (int k = 0; k < HMM_K; k += 4) {
            v0 = fmaf(bcast_lane(p, k + 0), Acol[k + 0], v0);
            v1 = fmaf(bcast_lane(p, k + 1), Acol[k + 1], v1);
            v2 = fmaf(bcast_lane(p, k + 2), Acol[k + 2], v2);
            v3 = fmaf(bcast_lane(p, k + 3), Acol[k + 3], v3);
        }
        const float v = (v0 + v1) + (v2 + v3);

        alpha = e + m + __logf(v);
    }

    // logsumexp over states, lane 0 writes per-batch result
    float m2 = alpha;
    #pragma unroll
    for (int off = 16; off; off >>= 1) m2 = fmaxf(m2, __shfl_xor(m2, off, 32));
    float s2 = __expf(alpha - m2);
    #pragma unroll
    for (int off = 16; off; off >>= 1) s2 += __shfl_xor(s2, off, 32);
    if (j == 0) res[b] = m2 + __logf(s2);
}

// ---------------------------------------------------------------------------
// Kernel 4: deterministic fixed-order sum of the 64 per-batch results.
// ---------------------------------------------------------------------------
__global__ void hmm_final(const float* __restrict__ res, float* __restrict__ out) {
    if (threadIdx.x == 0 && blockIdx.x == 0) {
        float s = 0.f;
        for (int b = 0; b < HMM_B; b++) s += res[b];
        out[0] = s;
    }
}

extern "C" void kernel_launch(void* const* d_in, const int* in_sizes, int n_in,
                              void* d_out, int out_size, void* d_ws, size_t ws_size,
                              hipStream_t stream) {
    const float* X     = (const float*)d_in[0];
    const float* sp    = (const float*)d_in[1];
    const float* trans = (const float*)d_in[2];
    const float* mu    = (const float*)d_in[3];
    const float* lv    = (const float*)d_in[4];
    float* ws  = (float*)d_ws;
    float* em  = ws + OFF_EM;
    float* out = (float*)d_out;

    hmm_prep<<<1, 32, 0, stream>>>(sp, trans, mu, lv, ws);

    const int tiles = (HMM_B * HMM_T) / 16;          // 16384 M-tiles
    hmm_emission<<<tiles / 8, 256, 0, stream>>>(X, ws, em);

    hmm_scan<<<HMM_B, 32, 0, stream>>>(ws, em, ws + OFF_RES);

    hmm_final<<<1, 1, 0, stream>>>(ws + OFF_RES, out);
}